// ReinforceLoss_29566554866422
// MI455X (gfx1250) — compile-verified
//
#include <hip/hip_runtime.h>
#include <hip/hip_bf16.h>
#include <math.h>
#include <stdint.h>

#define B_    64
#define N_    1001
#define NC_   1000
#define K_    10
#define MAXR_ 1024

#define AS1 __attribute__((address_space(1)))
#define AS3 __attribute__((address_space(3)))

typedef __attribute__((ext_vector_type(2))) float v2f;
typedef __attribute__((ext_vector_type(8))) float v8f;

#if __has_builtin(__builtin_amdgcn_global_load_async_to_lds_b32)
#define HAVE_ASYNC_LDS 1
#else
#define HAVE_ASYNC_LDS 0
#endif

// async global->LDS copy of one DWORD (per-lane addresses, ASYNCcnt tracked)
__device__ __forceinline__ void async_copy_b32(void* lds, const void* gmem) {
#if HAVE_ASYNC_LDS
  __builtin_amdgcn_global_load_async_to_lds_b32(
      (AS1 int*)(gmem), (AS3 int*)(lds), 0, 0);
#else
  *(AS3 int*)(lds) = *(const AS1 int*)(gmem);
#endif
}

__device__ __forceinline__ void async_wait0() {
#if __has_builtin(__builtin_amdgcn_s_wait_asynccnt)
  __builtin_amdgcn_s_wait_asynccnt(0);
#elif HAVE_ASYNC_LDS
  asm volatile("s_wait_asynccnt 0" ::: "memory");
#endif
}

// ---------------- wave32 helpers ----------------
__device__ __forceinline__ float wave_sum32(float v) {
  #pragma unroll
  for (int m = 16; m > 0; m >>= 1) v += __shfl_xor(v, m, 32);
  return v;
}
__device__ __forceinline__ int wave_max32(int v) {
  #pragma unroll
  for (int m = 16; m > 0; m >>= 1) v = max(v, __shfl_xor(v, m, 32));
  return v;
}

__device__ __forceinline__ uint32_t hash3(uint32_t a, uint32_t b, uint32_t c) {
  uint32_t h = a * 0x9E3779B1u ^ b * 0x85EBCA77u ^ c * 0xC2B2AE3Du;
  h ^= h >> 16; h *= 0x7FEB352Du;
  h ^= h >> 15; h *= 0x846CA68Bu;
  h ^= h >> 16;
  return h;
}

// ---------------- K1: log-softmax + Gumbel-max categorical sampling --------
__global__ void k_assign(const float* __restrict__ logits,
                         int* __restrict__ assign, float* __restrict__ lp_out) {
  int b = blockIdx.x;
  const float* lg = logits + (size_t)b * N_ * K_;
  float lps = 0.f;
  for (int i = threadIdx.x; i < NC_; i += blockDim.x) {
    const float* row = lg + (size_t)(i + 1) * K_;
    if (i + (int)blockDim.x < NC_)
      __builtin_prefetch(row + (size_t)K_ * blockDim.x, 0, 0);  // global_prefetch_b8
    float l[K_];
    float mx = -INFINITY;
    #pragma unroll
    for (int k = 0; k < K_; ++k) { l[k] = row[k]; mx = fmaxf(mx, l[k]); }
    float se = 0.f;
    #pragma unroll
    for (int k = 0; k < K_; ++k) se += expf(l[k] - mx);
    float lse = mx + logf(se);
    int a = 0; float gb = -INFINITY;
    #pragma unroll
    for (int k = 0; k < K_; ++k) {
      uint32_t h = hash3((uint32_t)b, (uint32_t)i, (uint32_t)k);
      float u = (float)(h >> 8) * (1.0f / 16777216.0f) + 1e-12f;
      float g = l[k] - logf(-logf(u));
      bool better = g > gb;
      gb = better ? g : gb;
      a = better ? k : a;
    }
    assign[(size_t)b * NC_ + i] = a;
    lps += l[a] - lse;
  }
  __shared__ float red[32];
  float s = wave_sum32(lps);
  int lane = threadIdx.x & 31, wid = threadIdx.x >> 5;
  if (lane == 0) red[wid] = s;
  __syncthreads();
  if (wid == 0) {
    int nw = blockDim.x >> 5;
    float v = (lane < nw) ? red[lane] : 0.f;
    v = wave_sum32(v);
    if (lane == 0) lp_out[b] = v;
  }
}

// ---------------- K2: capacity split scan -> contiguous route member lists -
__global__ void k_split(const int* __restrict__ assign,
                        const float* __restrict__ demands,
                        const float* __restrict__ capacity,
                        int* __restrict__ member, int* __restrict__ routeoff,
                        int* __restrict__ nroutes) {
  int b = blockIdx.x;
  int lane = threadIdx.x & 31;
  const int* as = assign + (size_t)b * NC_;
  const float* dm = demands + (size_t)b * N_;
  float cap = capacity[b];

  // pass 1: per-cluster member count + route count
  int nk = 0, routes = 0;
  if (lane < K_) {
    float load = 0.f; int cnt = 0; bool ne = false;
    for (int i = 0; i < NC_; ++i) {
      if (as[i] == lane) {
        float d = dm[i + 1];
        bool sn = (load + d > cap) && ne;
        cnt += sn ? 1 : 0;
        load = sn ? d : load + d;
        ne = true;
        nk++;
      }
    }
    routes = (nk > 0) ? (cnt + 1) : 0;
  }
  // exclusive prefix over lanes
  int cb = nk, rb = routes;
  #pragma unroll
  for (int off = 1; off < 32; off <<= 1) {
    int t = __shfl_up(cb, off, 32);
    int t2 = __shfl_up(rb, off, 32);
    if (lane >= off) { cb += t; rb += t2; }
  }
  int cluster_base = cb - nk;
  int route_base = rb - routes;
  int R = __shfl(rb, 31, 32);

  int* mem = member + (size_t)b * NC_;
  int* ro = routeoff + (size_t)b * (MAXR_ + 1);

  // pass 2: emit members (grouped by cluster, index order) + route starts
  if (lane < K_ && nk > 0) {
    float load = 0.f; int cnt = 0; bool ne = false; int c = 0;
    for (int i = 0; i < NC_; ++i) {
      if (as[i] == lane) {
        float d = dm[i + 1];
        bool sn = (load + d > cap) && ne;
        if (sn) cnt++;
        load = sn ? d : load + d;
        if (!ne || sn) ro[route_base + cnt] = cluster_base + c;
        ne = true;
        mem[cluster_base + c] = i;
        c++;
      }
    }
  }
  if (lane == 0) { ro[R] = NC_; nroutes[b] = R; }
}

// ---------------- K3: NN routing; 16 routes per wave, WMMA fp32 16x16x4 ----
// dist^2(c,p) = [cx, cy, |c|^2, 1] . [-2px, -2py, 1, |p|^2]  (A 16x4, B 4x16)
__global__ __launch_bounds__(512) void k_route(const float* __restrict__ coords,
                                               const int* __restrict__ member,
                                               const int* __restrict__ routeoff,
                                               const int* __restrict__ nroutes,
                                               float* __restrict__ dist_out) {
  __shared__ float scoord[2 * N_];       // raw (x,y) incl. depot, async-staged
  __shared__ float sx[NC_], sy[NC_], sa[NC_];
  __shared__ int vis[NC_];
  __shared__ int smem[NC_];
  __shared__ int sro[MAXR_ + 1];
  __shared__ float bacc;

  int b = blockIdx.x;
  const float* cbase = coords + (size_t)b * N_ * 2;

  // ---- async stage: coords / member / route offsets -> LDS (ASYNCcnt) ----
  for (int i = threadIdx.x; i < 2 * N_; i += blockDim.x)
    async_copy_b32(&scoord[i], &cbase[i]);
  for (int i = threadIdx.x; i < NC_; i += blockDim.x)
    async_copy_b32(&smem[i], &member[(size_t)b * NC_ + i]);
  for (int i = threadIdx.x; i < MAXR_ + 1; i += blockDim.x)
    async_copy_b32(&sro[i], &routeoff[(size_t)b * (MAXR_ + 1) + i]);
  if (threadIdx.x == 0) bacc = 0.f;
  async_wait0();                 // each wave drains its own ASYNCcnt
  __syncthreads();               // then cross-wave visibility

  // derive per-customer (x, y, |c|^2) and clear visited
  for (int i = threadIdx.x; i < NC_; i += blockDim.x) {
    float x = scoord[(i + 1) * 2], y = scoord[(i + 1) * 2 + 1];
    sx[i] = x; sy[i] = y; sa[i] = x * x + y * y;
    vis[i] = 0;
  }
  __syncthreads();

  float dpx = scoord[0], dpy = scoord[1];
  int R = nroutes[b];
  int lane = threadIdx.x & 31;
  int wid = threadIdx.x >> 5;
  int nwaves = blockDim.x >> 5;
  int ntiles = (R + 15) >> 4;

  for (int t = wid; t < ntiles; t += nwaves) {
    int r16 = t << 4;
    int col = lane & 15;
    int r = r16 + col;
    int roff = (r < R) ? sro[r] : NC_;
    int rend = (r < R) ? sro[r + 1] : NC_;
    int rlen = rend - roff;
    int roffc = __shfl(roff, col, 32);   // column-owner's [off,end)
    int rendc = __shfl(rend, col, 32);
    int maxlen = wave_max32(rlen);
    int c0 = __shfl(roff, 0, 32);        // tile member range start
    int cend = sro[min(r16 + 16, R)];    // tile member range end
    int nct = (cend - c0 + 15) >> 4;
    int rowbase = (lane >> 4) << 3;      // C layout: lanes16-31 hold rows 8..15
    bool lo = lane < 16;

    float px = dpx, py = dpy;            // lanes 0..15: route position
    float acc = 0.f;

    for (int s = 0; s < maxlen; ++s) {   // wave-uniform trip count (EXEC all-1s at WMMA)
      float pxo = __shfl(px, col, 32);
      float pyo = __shfl(py, col, 32);
      // B 4x16: lanes0-15 -> K0,K1 of col; lanes16-31 -> K2,K3 of col
      v2f bm;
      bm.x = lo ? (-2.f * px) : 1.f;
      bm.y = lo ? (-2.f * py) : (pxo * pxo + pyo * pyo);

      float best = INFINITY; int bestp = -1;
      for (int ct = 0; ct < nct; ++ct) {
        int mbase = c0 + (ct << 4);
        // A 16x4: lanes0-15 -> K0,K1 of row; lanes16-31 -> K2,K3 of row
        int ci = smem[min(mbase + col, NC_ - 1)];
        float vx = sx[ci], vy = sy[ci], va = sa[ci];   // unconditional -> cndmask
        v2f am;
        am.x = lo ? vx : va;
        am.y = lo ? vy : 1.f;
        v8f cz = {};
        v8f d2 = __builtin_amdgcn_wmma_f32_16x16x4_f32(
            false, am, false, bm, (short)0, cz, false, false);
        #pragma unroll
        for (int vr = 0; vr < 8; ++vr) {
          int pm = mbase + rowbase + vr;
          int cid = smem[min(pm, NC_ - 1)];
          int vv = vis[cid];                            // ds_load, branchless
          float dd = d2[vr];
          bool ok = (pm >= roffc) & (pm < rendc) & (vv == 0) & (dd < best);
          best = ok ? dd : best;
          bestp = ok ? pm : bestp;
        }
      }
      // combine the two half-column partials (lane ^ 16 holds rows 8..15)
      float ob = __shfl_xor(best, 16, 32);
      int op = __shfl_xor(bestp, 16, 32);
      bool tk = ob < best;
      best = tk ? ob : best;
      bestp = tk ? op : bestp;

      bool have = bestp >= 0;
      int ci = smem[have ? bestp : 0];
      if (lo & have) vis[ci] = 1;        // disjoint routes -> race-free
      float nx = sx[ci], ny = sy[ci];
      acc += have ? __builtin_sqrtf(fmaxf(best, 0.f)) : 0.f;
      px = have ? nx : px;
      py = have ? ny : py;
      asm volatile("s_wait_dscnt 0" ::: "memory");  // visited RAW across steps
    }
    // return-to-depot leg; lanes>=16 mirror lanes 0..15 -> count once
    float dx = px - dpx, dy = py - dpy;
    float closing = __builtin_sqrtf(dx * dx + dy * dy);
    acc = (lo & (rlen > 0)) ? (acc + closing) : (lo ? acc : 0.f);
    float wsum = wave_sum32(acc);
    if (lane == 0) atomicAdd(&bacc, wsum);
  }
  __syncthreads();
  if (threadIdx.x == 0) dist_out[b] = bacc;
}

// ---------------- K4: REINFORCE loss reduction -----------------------------
__global__ void k_loss(const float* __restrict__ lp, const float* __restrict__ dist,
                       float* __restrict__ out) {
  int tid = threadIdx.x;  // 64 threads
  float d = dist[tid], l = lp[tid];
  __shared__ float red[2], red2[2];
  int lane = tid & 31, wid = tid >> 5;
  float s = wave_sum32(d);
  if (lane == 0) red[wid] = s;
  __syncthreads();
  float mean = (red[0] + red[1]) * (1.0f / B_);
  float s2 = wave_sum32((d - mean) * l);
  if (lane == 0) red2[wid] = s2;
  __syncthreads();
  if (tid == 0) {
    out[0] = (red2[0] + red2[1]) * (1.0f / B_);  // /(B * n_samples)
    out[1] = mean;
  }
}

// ---------------- host launcher --------------------------------------------
extern "C" void kernel_launch(void* const* d_in, const int* in_sizes, int n_in,
                              void* d_out, int out_size, void* d_ws, size_t ws_size,
                              hipStream_t stream) {
  const float* logits = (const float*)d_in[0];
  const float* coords = (const float*)d_in[1];
  const float* demands = (const float*)d_in[2];
  const float* capacity = (const float*)d_in[3];
  (void)in_sizes; (void)n_in; (void)out_size; (void)ws_size;

  int* wsi = (int*)d_ws;
  int* assign = wsi;                                  // B*NC
  int* member = wsi + (size_t)B_ * NC_;               // B*NC
  int* routeoff = wsi + (size_t)2 * B_ * NC_;         // B*(MAXR+1)
  int* nroutes = routeoff + (size_t)B_ * (MAXR_ + 1); // B
  float* lp = (float*)(nroutes + B_);                 // B
  float* dist = lp + B_;                              // B

  k_assign<<<B_, 256, 0, stream>>>(logits, assign, lp);
  k_split<<<B_, 32, 0, stream>>>(assign, demands, capacity, member, routeoff, nroutes);
  k_route<<<B_, 512, 0, stream>>>(coords, member, routeoff, nroutes, dist);
  k_loss<<<1, 64, 0, stream>>>(lp, dist, (float*)d_out);
}